// TokensChooseScatterRouter_68315749810827
// MI455X (gfx1250) — compile-verified
//
#include <hip/hip_runtime.h>
#include <hip/hip_bf16.h>

// ---------------------------------------------------------------------------
// MoE tokens-choose router for MI455X (gfx1250, wave32).
// Roofline: GEMM reads 256MB f32 tokens @23.3TB/s ~= 11us floor. Router
// weights W (512KB) were being re-streamed per wave (8x256 blocks ~ 1GB L2
// traffic) -> now staged per-block into LDS by the Tensor Data Mover
// (TENSOR_LOAD_TO_LDS, double buffered, s_wait_tensorcnt), read back as
// LDS-broadcast B fragments. Matrix math: V_WMMA_F32_16X16X4_F32 (full f32 —
// routing is comparison-sensitive).
// ---------------------------------------------------------------------------

typedef __attribute__((ext_vector_type(2))) float v2f;
typedef __attribute__((ext_vector_type(8))) float v8f;
typedef __attribute__((ext_vector_type(4))) unsigned v4u;
typedef __attribute__((ext_vector_type(8))) int v8i;
typedef __attribute__((ext_vector_type(4))) int v4i;

constexpr int G = 8, T = 4096, H = 2048, E = 64, K = 2;
constexpr int TPB = 128;                    // tokens per block (8 waves x 16)
constexpr int BPG = T / TPB;                // 32 blocks per group
constexpr int NBLK1 = G * T / TPB;          // 256 GEMM blocks
constexpr int NCLAIM = K * T;               // 8192 claims per group
constexpr int SLABK = 128;                  // W k-slab rows staged per TDM op
constexpr int NS = H / SLABK;               // 16 slabs
constexpr int SLAB_ELEMS = SLABK * E;       // 8192 floats = 32KB

// ---------------------------------------------------------------------------
// K0: zero the per-(g,e) selection counters (int atomics -> deterministic).
// ---------------------------------------------------------------------------
__global__ void k_init(int* __restrict__ counts) {
  int i = blockIdx.x * blockDim.x + threadIdx.x;
  if (i < G * E) counts[i] = 0;
}

// TDM descriptor build + issue: copy SLAB_ELEMS contiguous f32 from W into
// LDS at lds_addr. D# per CDNA5 ISA:
//  group0: [1:0]=count=1 | [63:32]=lds_addr | [120:64]=global_addr | type=2
//  group1: wg_mask=0, data_size=2(4B), tensor_dim0=tile_dim0=SLAB_ELEMS,
//          tensor_dim1=tile_dim1=1, stride0=SLAB_ELEMS; groups 2/3 = 0 (2D).
// This toolchain's builtin takes 6 operands (g0, g1, g2, g3, extra, cpol).
__device__ __forceinline__ void tdm_load_w_slab(const float* __restrict__ W,
                                                unsigned lds_addr, int slab) {
  const unsigned long long ga =
      (unsigned long long)(size_t)(W + (size_t)slab * SLAB_ELEMS);
  v4u g0;
  g0.x = 1u;                                          // count=1, no gather
  g0.y = lds_addr;                                    // LDS byte address
  g0.z = (unsigned)ga;                                // global_addr[31:0]
  g0.w = (unsigned)((ga >> 32) & 0x01FFFFFFu) | (2u << 30);  // [56:32]|type=2
  v8i g1;
  g1[0] = (2 << 16);                                  // wg_mask=0, 4B elems
  g1[1] = (int)(((unsigned)SLAB_ELEMS & 0xFFFFu) << 16);      // dim0 lo
  g1[2] = (int)((((unsigned)SLAB_ELEMS >> 16) & 0xFFFFu) | (1u << 16));
  g1[3] = (int)(((unsigned)SLAB_ELEMS & 0xFFFFu) << 16);      // tile_dim0
  g1[4] = 1;                                          // tile_dim1=1
  g1[5] = SLAB_ELEMS;                                 // dim0 stride lo
  g1[6] = 0;
  g1[7] = 0;
  const v4i gz4 = {0, 0, 0, 0};
  const v8i gz8 = {0, 0, 0, 0, 0, 0, 0, 0};
  __builtin_amdgcn_tensor_load_to_lds(g0, g1, gz4, gz4, gz8, 0);
}

// ---------------------------------------------------------------------------
// K1: router GEMM (WMMA f32 16x16x4) + softmax + top-2 + partial reductions.
// One wave owns a 16-token x 64-expert tile: 4 accumulators of v8f.
// A fragment (16x4 f32): lane L<16 holds M=L,K={0,1}; lane L+16 holds K={2,3}
//   -> contiguous float2 global load per lane (HBM-bound stream).
// B fragments come from the TDM-staged LDS slab (double buffered).
// ---------------------------------------------------------------------------
__global__ __launch_bounds__(256) void k_router_gemm(
    const float* __restrict__ A, const float* __restrict__ W,
    const float* __restrict__ bias,
    float* __restrict__ weights, int* __restrict__ experts,
    float* __restrict__ pprobs, float* __restrict__ pz,
    int* __restrict__ counts) {
  __shared__ float Wl[2][SLAB_ELEMS];      // 64KB W staging (TDM dest)
  __shared__ float lds_logits[TPB * 65];   // stride 65 -> conflict-free cols
  __shared__ float lds_ms[TPB][2];         // per-token (max, sumexp)
  __shared__ float lds_z[TPB];             // per-token log_z^2

  const int tid  = threadIdx.x;
  const int wave = tid >> 5, lane = tid & 31;
  const int m    = lane & 15, kh = lane >> 4;

  const int gt0 = blockIdx.x * TPB;
  const int g   = gt0 / T;
  const int t0  = gt0 % T;
  const int tokBase = t0 + wave * 16;

  const float* Arow = A + ((size_t)g * T + tokBase + m) * H;
  const unsigned lds_w_base = (unsigned)(size_t)&Wl[0][0];

  v8f acc0 = {}, acc1 = {}, acc2 = {}, acc3 = {};

  if (tid < 32) tdm_load_w_slab(W, lds_w_base, 0);   // prologue: slab 0 -> buf 0

  for (int s = 0; s < NS; ++s) {
    const int cur = s & 1;
    if (tid < 32) {
      if (s + 1 < NS) {
        tdm_load_w_slab(W, lds_w_base + (unsigned)((cur ^ 1) * SLAB_ELEMS * 4),
                        s + 1);
        __builtin_amdgcn_s_wait_tensorcnt(1);   // slab s landed
      } else {
        __builtin_amdgcn_s_wait_tensorcnt(0);
      }
    }
    __syncthreads();                            // slab s visible to all waves

    const float* Wc = &Wl[cur][0];
    const float* Ak = Arow + s * SLABK;
    for (int kl = 0; kl < SLABK; kl += 4) {
      const float2 av = *reinterpret_cast<const float2*>(Ak + kl + 2 * kh);
      v2f a; a.x = av.x; a.y = av.y;
      const float* Wr0 = Wc + (kl + 2 * kh) * E;   // LDS row k
      v2f b0; b0.x = Wr0[ 0 + m]; b0.y = Wr0[E +  0 + m];
      v2f b1; b1.x = Wr0[16 + m]; b1.y = Wr0[E + 16 + m];
      v2f b2; b2.x = Wr0[32 + m]; b2.y = Wr0[E + 32 + m];
      v2f b3; b3.x = Wr0[48 + m]; b3.y = Wr0[E + 48 + m];
      acc0 = __builtin_amdgcn_wmma_f32_16x16x4_f32(false, a, false, b0,
                                                   (short)0, acc0, false, false);
      acc1 = __builtin_amdgcn_wmma_f32_16x16x4_f32(false, a, false, b1,
                                                   (short)0, acc1, false, false);
      acc2 = __builtin_amdgcn_wmma_f32_16x16x4_f32(false, a, false, b2,
                                                   (short)0, acc2, false, false);
      acc3 = __builtin_amdgcn_wmma_f32_16x16x4_f32(false, a, false, b3,
                                                   (short)0, acc3, false, false);
    }
    __syncthreads();   // everyone done with buf `cur` before TDM reuses it
  }

  // Spill logits (+bias) to LDS. C/D layout: VGPR r = row r (lanes 0-15) or
  // row r+8 (lanes 16-31), N = lane&15 within each 16-wide tile.
#pragma unroll
  for (int r = 0; r < 8; ++r) {
    const int row = wave * 16 + r + 8 * kh;
    lds_logits[row * 65 + ( 0 + m)] = acc0[r] + bias[ 0 + m];
    lds_logits[row * 65 + (16 + m)] = acc1[r] + bias[16 + m];
    lds_logits[row * 65 + (32 + m)] = acc2[r] + bias[32 + m];
    lds_logits[row * 65 + (48 + m)] = acc3[r] + bias[48 + m];
  }
  __syncthreads();

  // Per-token softmax + top-2 (E=64): one thread per token.
  if (tid < TPB) {
    const float* lr = &lds_logits[tid * 65];
    float mx = -3.402823466e+38f;
    float v1 = mx, v2 = mx; int i1 = 0, i2 = 0;
    for (int e = 0; e < E; ++e) {
      const float l = lr[e];
      mx = fmaxf(mx, l);
      if (l > v1)      { v2 = v1; i2 = i1; v1 = l; i1 = e; }
      else if (l > v2) { v2 = l; i2 = e; }
    }
    float s = 0.f;
    for (int e = 0; e < E; ++e) s += expf(lr[e] - mx);
    const float logz = mx + logf(s);
    lds_ms[tid][0] = mx;
    lds_ms[tid][1] = s;
    lds_z[tid] = logz * logz;
    const size_t base = ((size_t)g * T + t0 + tid) * K;
    weights[base + 0] = expf(v1 - mx) / s;
    weights[base + 1] = expf(v2 - mx) / s;
    experts[base + 0] = i1;
    experts[base + 1] = i2;
    atomicAdd(&counts[g * E + i1], 1);   // top-k indices distinct -> mask sum
    atomicAdd(&counts[g * E + i2], 1);
  }
  __syncthreads();

  // Deterministic partial prob-sums per expert for the aux loss.
  if (tid < E) {
    float sp = 0.f;
    for (int t = 0; t < TPB; ++t)
      sp += expf(lds_logits[t * 65 + tid] - lds_ms[t][0]) / lds_ms[t][1];
    pprobs[((size_t)g * BPG + (t0 / TPB)) * E + tid] = sp;
  }
  if (tid == 0) {
    float zs = 0.f;
    for (int t = 0; t < TPB; ++t) zs += lds_z[t];
    pz[blockIdx.x] = zs;
  }
}

// ---------------------------------------------------------------------------
// K2: rank of each token under stable descending sort by top-1 gate.
// rank(t) = #{j : w[j] > w[t]} + #{j < t : w[j] == w[t]}  (argsort(-w) ties).
// LDS-broadcast counting: 4096^2 cmp per group, spread over 8 blocks.
// ---------------------------------------------------------------------------
__global__ __launch_bounds__(512) void k_rank(const float* __restrict__ weights,
                                              int* __restrict__ rank) {
  __shared__ float w[T];   // 16KB
  const int g = blockIdx.x >> 3;
  const int chunk = blockIdx.x & 7;
  for (int i = threadIdx.x; i < T; i += 512)
    w[i] = weights[((size_t)g * T + i) * K];   // top-1 gate
  __syncthreads();
  const int t = chunk * 512 + threadIdx.x;
  const float wt = w[t];
  int r = 0;
  for (int j = 0; j < T; ++j) {
    const float wj = w[j];
    r += (wj > wt) || (wj == wt && j < t);
  }
  rank[g * T + t] = r;
}

// ---------------------------------------------------------------------------
// K3: scatter claim experts into flattened sorted order: pos = k*T + rank(t).
// ---------------------------------------------------------------------------
__global__ void k_scatter(const int* __restrict__ experts,
                          const int* __restrict__ rank,
                          int* __restrict__ sorted_e) {
  const int idx = blockIdx.x * blockDim.x + threadIdx.x;
  if (idx >= G * T * K) return;
  const int k = idx & (K - 1);
  const int gt = idx >> 1;
  const int g = gt / T, t = gt % T;
  sorted_e[(size_t)g * NCLAIM + k * T + rank[g * T + t]] = experts[idx];
}

// ---------------------------------------------------------------------------
// K4: per-expert exclusive claim counts in position order (== cumsum-1).
// Deterministic: wave32 ballots build per-wave per-expert histograms + exact
// in-wave ranks; cross-wave/chunk prefixes carried in LDS. No atomics.
// ---------------------------------------------------------------------------
__global__ __launch_bounds__(256) void k_prio(const int* __restrict__ sorted_e,
                                              int* __restrict__ sorted_p) {
  __shared__ int hist[8][E];
  __shared__ int pref[8][E];
  __shared__ int base[E];
  const int g = blockIdx.x;
  const int tid = threadIdx.x, wave = tid >> 5, lane = tid & 31;
  const unsigned below = (1u << lane) - 1u;
  if (tid < E) base[tid] = 0;
  __syncthreads();
  for (int c = 0; c < NCLAIM / 256; ++c) {
    const int p = c * 256 + tid;
    const int e = sorted_e[(size_t)g * NCLAIM + p];
    int lane_rank = 0;
    for (int ee = 0; ee < E; ++ee) {
      const unsigned m32 = (unsigned)__ballot(e == ee);
      if (lane == 0) hist[wave][ee] = __popc(m32);
      if (e == ee) lane_rank = __popc(m32 & below);
    }
    __syncthreads();
    if (tid < E) {
      int s = base[tid];
#pragma unroll
      for (int w = 0; w < 8; ++w) { pref[w][tid] = s; s += hist[w][tid]; }
      base[tid] = s;
    }
    __syncthreads();
    sorted_p[(size_t)g * NCLAIM + p] = pref[wave][e] + lane_rank;
    __syncthreads();
  }
}

// ---------------------------------------------------------------------------
// K5: gather priorities back to original token order, apply capacity mask,
// emit dispatch_indices [G,T,K,2] and combine_weights [G,T,K].
// ---------------------------------------------------------------------------
__global__ void k_final(const float* __restrict__ weights,
                        const int* __restrict__ experts,
                        const int* __restrict__ rank,
                        const int* __restrict__ sorted_p,
                        const int* __restrict__ capacity,
                        float* __restrict__ out) {
  const int idx = blockIdx.x * blockDim.x + threadIdx.x;
  if (idx >= G * T * K) return;
  const int k = idx & (K - 1);
  const int gt = idx >> 1;
  const int g = gt / T, t = gt % T;
  const int prio = sorted_p[(size_t)g * NCLAIM + k * T + rank[g * T + t]];
  out[(size_t)idx * 2 + 0] = (float)experts[idx];
  out[(size_t)idx * 2 + 1] = (float)prio;
  out[(size_t)G * T * K * 2 + idx] =
      weights[idx] * ((prio < capacity[0]) ? 1.f : 0.f);
}

// ---------------------------------------------------------------------------
// K6: deterministic loss reductions.
// aux = mean_{g,e}[(counts/T)*(sum_probs/T)] * E^2 ; z = sum(log_z^2)/(G*T).
// ---------------------------------------------------------------------------
__global__ __launch_bounds__(512) void k_losses(const float* __restrict__ pprobs,
                                                const float* __restrict__ pz,
                                                const int* __restrict__ counts,
                                                float* __restrict__ out) {
  __shared__ float red[512];
  const int tid = threadIdx.x;
  const int g = tid >> 6, e = tid & 63;
  float sp = 0.f;
  for (int b = 0; b < BPG; ++b) sp += pprobs[((size_t)g * BPG + b) * E + e];
  red[tid] = ((float)counts[g * E + e] / (float)T) * (sp / (float)T);
  __syncthreads();
  for (int s = 256; s > 0; s >>= 1) {
    if (tid < s) red[tid] += red[tid + s];
    __syncthreads();
  }
  if (tid == 0)
    out[(size_t)G * T * K * 3 + 0] =
        red[0] / (float)(G * E) * (float)(E * E);
  __syncthreads();
  red[tid] = (tid < NBLK1) ? pz[tid] : 0.f;
  __syncthreads();
  for (int s = 256; s > 0; s >>= 1) {
    if (tid < s) red[tid] += red[tid + s];
    __syncthreads();
  }
  if (tid == 0) out[(size_t)G * T * K * 3 + 1] = red[0] / (float)(G * T);
}

// ---------------------------------------------------------------------------
extern "C" void kernel_launch(void* const* d_in, const int* in_sizes, int n_in,
                              void* d_out, int out_size, void* d_ws,
                              size_t ws_size, hipStream_t stream) {
  const float* A    = (const float*)d_in[0];   // [G,T,H]
  const float* W    = (const float*)d_in[1];   // [H,E]
  const float* bias = (const float*)d_in[2];   // [E]
  const int* capacity = (const int*)d_in[4];   // scalar
  float* out = (float*)d_out;

  char* ws = (char*)d_ws;
  size_t off = 0;
  auto carve = [&](size_t bytes) {
    void* p = ws + off;
    off += (bytes + 255) & ~(size_t)255;
    return p;
  };
  float* weights  = (float*)carve((size_t)G * T * K * sizeof(float));
  int*   experts  = (int*)  carve((size_t)G * T * K * sizeof(int));
  int*   rank     = (int*)  carve((size_t)G * T * sizeof(int));
  int*   sorted_e = (int*)  carve((size_t)G * NCLAIM * sizeof(int));
  int*   sorted_p = (int*)  carve((size_t)G * NCLAIM * sizeof(int));
  float* pprobs   = (float*)carve((size_t)G * BPG * E * sizeof(float));
  float* pz       = (float*)carve((size_t)NBLK1 * sizeof(float));
  int*   counts   = (int*)  carve((size_t)G * E * sizeof(int));

  k_init<<<1, 512, 0, stream>>>(counts);
  k_router_gemm<<<NBLK1, 256, 0, stream>>>(A, W, bias, weights, experts,
                                           pprobs, pz, counts);
  k_rank<<<G * 8, 512, 0, stream>>>(weights, rank);
  k_scatter<<<(G * T * K) / 256, 256, 0, stream>>>(experts, rank, sorted_e);
  k_prio<<<G, 256, 0, stream>>>(sorted_e, sorted_p);
  k_final<<<(G * T * K) / 256, 256, 0, stream>>>(weights, experts, rank,
                                                 sorted_p, capacity, out);
  k_losses<<<1, 512, 0, stream>>>(pprobs, pz, counts, out);
}